// VectorQuantizer2_51951924412570
// MI455X (gfx1250) — compile-verified
//
#include <hip/hip_runtime.h>

typedef __attribute__((ext_vector_type(2))) float v2f;
typedef __attribute__((ext_vector_type(8))) float v8f;

#define BB 64
#define CC 32
#define HH 16
#define VV 4096
#define NPIX 256              // 16*16
#define FSIZE (BB*CC*NPIX)    // 524288

// ---------------------------------------------------------------- init
__global__ void vq_init_kernel(const float* __restrict__ f,
                               float* __restrict__ f_rest,
                               float* __restrict__ f_hat,
                               float* __restrict__ loss) {
    int tid = blockIdx.x * blockDim.x + threadIdx.x;
    if (tid < FSIZE) { f_rest[tid] = f[tid]; f_hat[tid] = 0.0f; }
    if (tid == 0) *loss = 0.0f;
}

// ------------------------------------------------------- 0.5 * ||e||^2
__global__ void vq_esq_kernel(const float* __restrict__ emb,
                              float* __restrict__ half_esq) {
    int v = blockIdx.x * blockDim.x + threadIdx.x;
    if (v >= VV) return;
    float s = 0.0f;
    #pragma unroll
    for (int c = 0; c < CC; ++c) { float e = emb[v*CC + c]; s += e*e; }
    half_esq[v] = 0.5f * s;
}

// ---------------------------------------------------------- area pool
// z[n][c], n = b*pn*pn + py*pn + px   (matches transpose(0,2,3,1).reshape)
__global__ void vq_pool_kernel(const float* __restrict__ f_rest,
                               float* __restrict__ z, int pn) {
    int tid = blockIdx.x * blockDim.x + threadIdx.x;
    int total = BB * pn * pn * CC;
    if (tid >= total) return;
    int c  = tid & (CC-1);
    int n  = tid >> 5;
    int px = n % pn;
    int py = (n / pn) % pn;
    int b  = n / (pn * pn);
    int k  = HH / pn;
    const float* base = f_rest + ((b*CC + c)*HH + py*k)*HH + px*k;
    float s = 0.0f;
    for (int yy = 0; yy < k; ++yy)
        for (int xx = 0; xx < k; ++xx)
            s += base[yy*HH + xx];
    z[n*CC + c] = s / (float)(k*k);
}

// -------------------------------------------- WMMA nearest-code search
// score = z.e - 0.5*||e||^2 ; argmax score == argmin L2 distance.
// One wave handles 16 rows; 256 tiles of 16 codes; K=32 as 8x k=4 WMMA.
__global__ void vq_argmin_wmma_kernel(const float* __restrict__ z,
                                      const float* __restrict__ emb,
                                      const float* __restrict__ half_esq,
                                      int* __restrict__ idx_out, int N) {
    int wave = threadIdx.x >> 5;
    int lane = threadIdx.x & 31;
    int m0 = (blockIdx.x * 4 + wave) * 16;
    if (m0 >= N) return;                       // wave-uniform guard
    int half = lane >> 4;                      // 0 | 1
    int col  = lane & 15;
    int koff = half * 2;                       // K lanes-16..31 hold K=2,3

    // A fragments for all 8 K-chunks (persist in VGPRs)
    v2f a[8];
    const float* zr = z + (m0 + col) * CC;
    #pragma unroll
    for (int k = 0; k < 8; ++k) {
        a[k].x = zr[4*k + koff];
        a[k].y = zr[4*k + koff + 1];
    }

    float best[8];
    int   bidx[8];
    #pragma unroll
    for (int r = 0; r < 8; ++r) { best[r] = -3.4e38f; bidx[r] = 0; }

    for (int v0 = 0; v0 < VV; v0 += 16) {
        v8f acc = {0.f,0.f,0.f,0.f,0.f,0.f,0.f,0.f};
        const float* er = emb + (v0 + col) * CC;
        #pragma unroll
        for (int k = 0; k < 8; ++k) {
            v2f bf;
            bf.x = er[4*k + koff];
            bf.y = er[4*k + koff + 1];
            acc = __builtin_amdgcn_wmma_f32_16x16x4_f32(
                    false, a[k], false, bf, (short)0, acc, false, false);
        }
        float he = half_esq[v0 + col];
        int vidx = v0 + col;
        #pragma unroll
        for (int r = 0; r < 8; ++r) {
            float s = acc[r] - he;
            if (s > best[r]) { best[r] = s; bidx[r] = vidx; }
        }
    }
    // argmax across the 16 lanes of each half (C/D layout: N = lane&15)
    #pragma unroll
    for (int r = 0; r < 8; ++r) {
        float v = best[r]; int bi = bidx[r];
        #pragma unroll
        for (int m = 8; m >= 1; m >>= 1) {
            float ov = __shfl_xor(v, m, 32);
            int   oi = __shfl_xor(bi, m, 32);
            if (ov > v || (ov == v && oi < bi)) { v = ov; bi = oi; }
        }
        if (col == 0) idx_out[m0 + half*8 + r] = bi;   // M = r + 8*half
    }
}

// ---------------------------------------- gather + bicubic upsample
__device__ __forceinline__ void cubic_weights(float t, float w[4]) {
    const float a = -0.5f;
    float t2 = t*t, t3 = t2*t;
    w[0] = a * (t3 - 2.0f*t2 + t);
    w[1] = 1.5f*t3 - 2.5f*t2 + 1.0f;
    float u = 1.0f - t, u2 = u*u, u3 = u2*u;
    w[2] = 1.5f*u3 - 2.5f*u2 + 1.0f;
    w[3] = a * (t2 - t3);
}

__global__ void vq_upsample_kernel(const int* __restrict__ idx,
                                   const float* __restrict__ emb,
                                   float* __restrict__ h_up, int pn) {
    int tid = blockIdx.x * blockDim.x + threadIdx.x;
    if (tid >= FSIZE) return;
    int x = tid & 15, y = (tid >> 4) & 15, c = (tid >> 8) & 31, b = tid >> 13;
    float scale = (float)pn / 16.0f;
    float fy = (y + 0.5f) * scale - 0.5f;
    float fx = (x + 0.5f) * scale - 0.5f;
    int iy = (int)floorf(fy), ix = (int)floorf(fx);
    float wy[4], wx[4];
    cubic_weights(fy - (float)iy, wy);
    cubic_weights(fx - (float)ix, wx);
    float s = 0.0f;
    #pragma unroll
    for (int j = 0; j < 4; ++j) {
        int py = min(max(iy - 1 + j, 0), pn - 1);
        float rs = 0.0f;
        #pragma unroll
        for (int i = 0; i < 4; ++i) {
            int px = min(max(ix - 1 + i, 0), pn - 1);
            int n = (b*pn + py)*pn + px;
            rs += wx[i] * emb[idx[n]*CC + c];
        }
        s += wy[j] * rs;
    }
    h_up[tid] = s;
}

// ------------------------- phi conv3x3 + residual update + loss partials
__global__ void vq_conv_update_kernel(const float* __restrict__ h_up,
                                      const float* __restrict__ w,    // [32,32,3,3]
                                      const float* __restrict__ bias, // [32]
                                      const float* __restrict__ f_in,
                                      float* __restrict__ f_hat,
                                      float* __restrict__ f_rest,
                                      float* __restrict__ partial) {
    __shared__ float sh[CC * NPIX];   // 32 KB activation tile for this b
    __shared__ float sw[CC * 9];
    __shared__ float red[NPIX];
    int b = blockIdx.x, co = blockIdx.y;
    int t = threadIdx.x;              // pixel 0..255
    for (int ci = 0; ci < CC; ++ci)
        sh[ci*NPIX + t] = h_up[(b*CC + ci)*NPIX + t];
    for (int i = t; i < CC*9; i += NPIX)
        sw[i] = w[co*CC*9 + i];
    __syncthreads();

    int x = t & 15, y = t >> 4;
    float acc = bias[co];
    for (int ci = 0; ci < CC; ++ci) {
        const float* sp = sh + ci*NPIX;
        const float* wp = sw + ci*9;
        #pragma unroll
        for (int ky = 0; ky < 3; ++ky) {
            int yy = y + ky - 1;
            if (yy < 0 || yy > 15) continue;
            #pragma unroll
            for (int kx = 0; kx < 3; ++kx) {
                int xx = x + kx - 1;
                if (xx < 0 || xx > 15) continue;
                acc += wp[ky*3 + kx] * sp[yy*16 + xx];
            }
        }
    }
    float hphi = 0.5f * sh[co*NPIX + t] + 0.5f * acc;
    int gi = (b*CC + co)*NPIX + t;
    float fh = f_hat[gi] + hphi;
    f_hat[gi] = fh;
    f_rest[gi] -= hphi;
    float d = fh - f_in[gi];
    red[t] = d * d;
    __syncthreads();
    for (int s = NPIX/2; s > 0; s >>= 1) {
        if (t < s) red[t] += red[t + s];
        __syncthreads();
    }
    if (t == 0) partial[b*CC + co] = red[0];
}

// loss += 0.25 * mean(sq)   (fixed-order deterministic reduction)
__global__ void vq_loss_reduce_kernel(const float* __restrict__ partial,
                                      float* __restrict__ loss) {
    __shared__ float red[256];
    int t = threadIdx.x;
    float s = 0.0f;
    for (int i = t; i < BB*CC; i += 256) s += partial[i];
    red[t] = s;
    __syncthreads();
    for (int d = 128; d > 0; d >>= 1) {
        if (t < d) red[t] += red[t + d];
        __syncthreads();
    }
    if (t == 0) *loss += 0.25f * red[0] / (float)FSIZE;
}

// ---------------------------------------------------------------- host
extern "C" void kernel_launch(void* const* d_in, const int* in_sizes, int n_in,
                              void* d_out, int out_size, void* d_ws, size_t ws_size,
                              hipStream_t stream) {
    const float* f     = (const float*)d_in[0];   // [64,32,16,16]
    const float* emb   = (const float*)d_in[1];   // [4096,32]
    const float* phi_w = (const float*)d_in[2];   // [4,32,32,3,3]
    const float* phi_b = (const float*)d_in[3];   // [4,32]

    float* f_hat = (float*)d_out;                 // first output, 524288 floats
    float* loss  = (float*)d_out + FSIZE;         // scalar loss

    float* ws       = (float*)d_ws;
    float* f_rest   = ws;                         // 524288
    float* z        = ws + FSIZE;                 // 524288
    float* h_up     = ws + 2*FSIZE;               // 524288
    float* half_esq = ws + 3*FSIZE;               // 4096
    int*   idx      = (int*)(ws + 3*FSIZE + VV);  // 16384
    float* partial  = ws + 3*FSIZE + VV + 16384;  // 2048

    vq_init_kernel<<<(FSIZE+255)/256, 256, 0, stream>>>(f, f_rest, f_hat, loss);
    vq_esq_kernel<<<VV/256, 256, 0, stream>>>(emb, half_esq);

    const int pns[5]     = {1, 2, 4, 8, 16};
    const int phi_idx[5] = {0, 1, 1, 2, 3};
    for (int si = 0; si < 5; ++si) {
        int pn = pns[si];
        int N  = BB * pn * pn;
        int pi = phi_idx[si];

        vq_pool_kernel<<<(N*CC + 255)/256, 256, 0, stream>>>(f_rest, z, pn);
        vq_argmin_wmma_kernel<<<N/64, 128, 0, stream>>>(z, emb, half_esq, idx, N);
        vq_upsample_kernel<<<FSIZE/256, 256, 0, stream>>>(idx, emb, h_up, pn);
        vq_conv_update_kernel<<<dim3(BB, CC), NPIX, 0, stream>>>(
            h_up, phi_w + pi*CC*CC*9, phi_b + pi*CC, f, f_hat, f_rest, partial);
        vq_loss_reduce_kernel<<<1, 256, 0, stream>>>(partial, loss);
    }
}